// SpeciesSpecificProjectionHead_28802050687054
// MI455X (gfx1250) — compile-verified
//
#include <hip/hip_runtime.h>
#include <math.h>

#define E_ 16
#define H_ 768
#define D_ 768
#define L_ 32
#define B_ 2048
#define S_ 256
#define TILES_ (B_ / 16)   // 128 row-tiles max per expert
#define AP_ 772            // padded A-tile row stride (bank-conflict avoidance)
#define KPANELS_ (H_ / 16) // 48 K-panels of 16

typedef __attribute__((ext_vector_type(2))) float v2f;
typedef __attribute__((ext_vector_type(8))) float v8f;
typedef __attribute__((ext_vector_type(4))) unsigned int v4u;
typedef __attribute__((ext_vector_type(8))) int v8i;
typedef __attribute__((ext_vector_type(4))) int v4i;

#if __has_builtin(__builtin_amdgcn_tensor_load_to_lds) && \
    __has_builtin(__builtin_amdgcn_s_wait_tensorcnt)
#define HAVE_TDM_ 1
#else
#define HAVE_TDM_ 0
#endif

// ---------------------------------------------------------------------------
// Kernel 1: mean-pool over S + LayerNorm (shared stats) + routed gamma/beta.
// Streams the dominant 1.6 GB with non-temporal loads (keep W1/W2 in L2).
// ---------------------------------------------------------------------------
__global__ __launch_bounds__(256) void pool_ln_affine(
    const float* __restrict__ hs, const int* __restrict__ sp,
    const float* __restrict__ gamma, const float* __restrict__ beta,
    float* __restrict__ xe) {
  const int b = blockIdx.x;
  const int tid = threadIdx.x;
  const float* base = hs + (size_t)b * S_ * H_;
  float a0 = 0.f, a1 = 0.f, a2 = 0.f;
  for (int s = 0; s < S_; ++s) {
    const float* row = base + s * H_;
    a0 += __builtin_nontemporal_load(row + tid);
    a1 += __builtin_nontemporal_load(row + tid + 256);
    a2 += __builtin_nontemporal_load(row + tid + 512);
  }
  const float inv_s = 1.0f / (float)S_;
  const float p0 = a0 * inv_s, p1 = a1 * inv_s, p2 = a2 * inv_s;

  __shared__ float rs[256], rq[256];
  rs[tid] = p0 + p1 + p2;
  rq[tid] = p0 * p0 + p1 * p1 + p2 * p2;
  __syncthreads();
  for (int off = 128; off > 0; off >>= 1) {
    if (tid < off) { rs[tid] += rs[tid + off]; rq[tid] += rq[tid + off]; }
    __syncthreads();
  }
  const float mu = rs[0] * (1.0f / (float)H_);
  const float var = rq[0] * (1.0f / (float)H_) - mu * mu;
  const float rstd = rsqrtf(var + 1e-5f);

  const int e = sp[b];
  const float* g = gamma + e * H_;
  const float* bt = beta + e * H_;
  float* o = xe + (size_t)b * H_;
  o[tid]       = (p0 - mu) * rstd * g[tid]       + bt[tid];
  o[tid + 256] = (p1 - mu) * rstd * g[tid + 256] + bt[tid + 256];
  o[tid + 512] = (p2 - mu) * rstd * g[tid + 512] + bt[tid + 512];
}

// ---------------------------------------------------------------------------
// Kernel 2: deterministic per-expert sample lists (ascending order, no atomics)
// ---------------------------------------------------------------------------
__global__ void build_lists(const int* __restrict__ sp,
                            int* __restrict__ counts, int* __restrict__ lists) {
  const int e = threadIdx.x;
  if (e < E_) {
    int c = 0;
    for (int b = 0; b < B_; ++b)
      if (sp[b] == e) lists[e * B_ + c++] = b;
    counts[e] = c;
  }
}

#if HAVE_TDM_
// ---------------------------------------------------------------------------
// TDM: DMA one W1 K-panel (16 rows x 768 f32, row stride 768) global -> LDS.
// D# per CDNA5 ISA: 2D tensor, data_size=4B, type=2,
// count=1, tile_dim0=768 (x, contiguous), tile_dim1=16 (y), dim0_stride=768.
// Groups 2/3 and the extra group are zero (<=2D tensor). Tracked by TENSORcnt.
// ---------------------------------------------------------------------------
__device__ __forceinline__ void tdm_load_w1_panel(const float* gsrc, float* ldst) {
  const unsigned long long ga = (unsigned long long)(uintptr_t)gsrc;
  const unsigned int ldsoff = (unsigned int)(uintptr_t)ldst;  // low 32b = LDS byte offset
  v4u g0;
  g0.x = 1u;                                                  // count=1, user descriptor
  g0.y = ldsoff;                                              // lds_addr
  g0.z = (unsigned int)ga;                                    // global_addr[31:0]
  g0.w = (unsigned int)((ga >> 32) & 0x01FFFFFFu) | 0x80000000u; // addr[56:32] | type=2
  v8i g1;
  g1[0] = (int)(2u << 16);          // workgroup_mask=0, data_size=2 (4 bytes)
  g1[1] = (int)((unsigned)D_ << 16);// tensor_dim0 = 768 (low 16 bits)
  g1[2] = (int)((unsigned)H_ << 16);// tensor_dim1 = 768 (low 16 bits)
  g1[3] = (int)((unsigned)D_ << 16);// tile_dim0 = 768
  g1[4] = 16;                       // tile_dim1 = 16, tile_dim2 = 0
  g1[5] = D_;                       // tensor_dim0_stride = 768 (low 32)
  g1[6] = 0;
  g1[7] = 0;
  const v4i z4 = {0, 0, 0, 0};
  const v8i z8 = {0, 0, 0, 0, 0, 0, 0, 0};
  __builtin_amdgcn_tensor_load_to_lds(g0, g1, z4, z4, z8, 0);
}
#endif

// ---------------------------------------------------------------------------
// Kernel 3: grouped per-expert MLP with fp32 WMMA.
// Block = (expert e, 16-row tile t), 8 waves. Dynamic LDS:
//   Als: 16 x AP_   A-tile (xe rows), reused for GELU(h)
//   P0 : 16 x 768   W1 K-panel buffer 0   (aliased by Yred after GEMM1)
//   P1 : 16 x 768   W1 K-panel buffer 1
// W1 panels are double-buffered via the Tensor Data Mover (wave 0 issues,
// s_wait_tensorcnt(1) overlaps panel k+1 DMA with panel k WMMA compute).
// ---------------------------------------------------------------------------
__global__ __launch_bounds__(256, 2) void expert_mlp(
    const float* __restrict__ xe,
    const float* __restrict__ W1, const float* __restrict__ b1,
    const float* __restrict__ W2, const float* __restrict__ b2,
    const int* __restrict__ counts, const int* __restrict__ lists,
    float* __restrict__ out) {
  extern __shared__ float smem[];
  float* Als = smem;                        // 16*AP_ floats
  float* P0  = smem + 16 * AP_;             // 16*768 floats
  float* P1  = P0 + 16 * D_;                // 16*768 floats
  float* Yred = P0;                         // aliases P0 (panels dead in GEMM2)

  const int e = blockIdx.x / TILES_;
  const int t = blockIdx.x % TILES_;
  const int cnt = counts[e];
  if (t * 16 >= cnt) return;                // uniform per block

  const int tid  = threadIdx.x;
  const int wave = tid >> 5;
  const int lane = tid & 31;
  const int half = lane >> 4;
  const int l16  = lane & 15;
  const int* rowlist = lists + e * B_ + t * 16;
  const float* W1e = W1 + (size_t)e * H_ * D_;

#if HAVE_TDM_
  if (wave == 0) tdm_load_w1_panel(W1e, P0);   // prefetch panel 0 during A staging
#endif

  // ---- stage A tile: 16 rows of xe (zero-pad past cnt) ----
  for (int j = 0; j < 12; ++j) {
    const int f  = tid + j * 256;
    const int r  = f / 192;
    const int c4 = f % 192;
    float4 v = make_float4(0.f, 0.f, 0.f, 0.f);
    if (t * 16 + r < cnt)
      v = *(const float4*)(xe + (size_t)rowlist[r] * H_ + c4 * 4);
    *(float4*)(Als + r * AP_ + c4 * 4) = v;
  }

  v8f acc[6] = {};

  // ---- GEMM1: (16 x 768) @ (768 x 768), double-buffered K-panels ----
  for (int kp = 0; kp < KPANELS_; ++kp) {
    const int k0 = kp * 16;
    float* Pcur = (kp & 1) ? P1 : P0;
    __syncthreads();   // everyone done reading the buffer about to be refilled
#if HAVE_TDM_
    if (wave == 0) {
      if (kp + 1 < KPANELS_) {
        tdm_load_w1_panel(W1e + (size_t)(k0 + 16) * D_, (kp & 1) ? P0 : P1);
        __builtin_amdgcn_s_wait_tensorcnt(1);  // panel kp complete (in-order)
      } else {
        __builtin_amdgcn_s_wait_tensorcnt(0);
      }
    }
#else
    for (int j = 0; j < 12; ++j) {
      const int f  = tid + j * 256;
      const int r  = f / 192;
      const int c4 = f % 192;
      *(float4*)(Pcur + r * D_ + c4 * 4) =
          *(const float4*)(W1e + (size_t)(k0 + r) * D_ + c4 * 4);
    }
#endif
    __syncthreads();   // panel kp visible to all waves

    v2f afrag[4];
    for (int kk = 0; kk < 4; ++kk) {
      const int kb = k0 + kk * 4 + 2 * half;   // A 16x4: v0={K0|K2}, v1={K1|K3}
      afrag[kk].x = Als[l16 * AP_ + kb];
      afrag[kk].y = Als[l16 * AP_ + kb + 1];
    }
    for (int nt = 0; nt < 6; ++nt) {
      const int n0 = (wave * 6 + nt) * 16;
      for (int kk = 0; kk < 4; ++kk) {
        const int kr = kk * 4 + 2 * half;      // B 4x16 mirrored layout
        v2f bfrag;
        bfrag.x = Pcur[kr * D_ + n0 + l16];
        bfrag.y = Pcur[(kr + 1) * D_ + n0 + l16];
        acc[nt] = __builtin_amdgcn_wmma_f32_16x16x4_f32(
            false, afrag[kk], false, bfrag, (short)0, acc[nt], false, false);
      }
    }
  }
  __syncthreads();

  // ---- bias + exact GELU, write h back into Als ----
  const float inv_sqrt2 = 0.70710678118654752f;
  for (int nt = 0; nt < 6; ++nt) {
    const int n = (wave * 6 + nt) * 16 + l16;
    const float bb = b1[e * D_ + n];
    for (int i = 0; i < 8; ++i) {
      const int m = half * 8 + i;              // C layout: VGPR i = rows i / 8+i
      const float x = acc[nt][i] + bb;
      Als[m * AP_ + n] = 0.5f * x * (1.0f + erff(x * inv_sqrt2));
    }
  }
  __syncthreads();

  // ---- GEMM2: (16 x 768) @ (768 x 32), K split across 8 waves ----
  const float* W2e = W2 + (size_t)e * D_ * L_;
  v8f acc2[2] = {};
  const int kstart = wave * 96;
  for (int ks = 0; ks < 24; ++ks) {
    const int kb = kstart + ks * 4 + 2 * half;
    v2f a2;
    a2.x = Als[l16 * AP_ + kb];
    a2.y = Als[l16 * AP_ + kb + 1];
    for (int nt = 0; nt < 2; ++nt) {
      const int n0 = nt * 16;
      v2f bf;
      bf.x = W2e[kb * L_ + n0 + l16];
      bf.y = W2e[(kb + 1) * L_ + n0 + l16];
      acc2[nt] = __builtin_amdgcn_wmma_f32_16x16x4_f32(
          false, a2, false, bf, (short)0, acc2[nt], false, false);
    }
  }
  for (int nt = 0; nt < 2; ++nt) {
    const int n = nt * 16 + l16;
    for (int i = 0; i < 8; ++i) {
      const int m = half * 8 + i;
      Yred[wave * 512 + m * L_ + n] = acc2[nt][i];
    }
  }
  __syncthreads();

  // ---- deterministic cross-wave reduction + bias + scatter to logits ----
  for (int q = 0; q < 2; ++q) {
    const int o = tid + q * 256;       // (m, n) in 16x32
    const int m = o >> 5;
    const int n = o & 31;
    if (t * 16 + m < cnt) {
      float v = 0.f;
      for (int w = 0; w < 8; ++w) v += Yred[w * 512 + o];
      v += b2[e * L_ + n];
      out[(size_t)rowlist[m] * L_ + n] = v;
    }
  }
}

// ---------------------------------------------------------------------------
extern "C" void kernel_launch(void* const* d_in, const int* in_sizes, int n_in,
                              void* d_out, int out_size, void* d_ws, size_t ws_size,
                              hipStream_t stream) {
  const float* hs    = (const float*)d_in[0];
  const int*   sp    = (const int*)d_in[1];
  const float* gamma = (const float*)d_in[2];
  const float* beta  = (const float*)d_in[3];
  const float* W1    = (const float*)d_in[4];
  const float* b1    = (const float*)d_in[5];
  const float* W2    = (const float*)d_in[6];
  const float* b2    = (const float*)d_in[7];
  float* out = (float*)d_out;

  float* xe     = (float*)d_ws;                  // B*H floats = 6.29 MB
  int*   counts = (int*)(xe + (size_t)B_ * H_);  // E ints
  int*   lists  = counts + E_;                   // E*B ints

  pool_ln_affine<<<B_, 256, 0, stream>>>(hs, sp, gamma, beta, xe);
  build_lists<<<1, 32, 0, stream>>>(sp, counts, lists);

  const size_t smem_bytes = (size_t)(16 * AP_ + 2 * 16 * D_) * sizeof(float);
  (void)hipFuncSetAttribute((const void*)expert_mlp,
                            hipFuncAttributeMaxDynamicSharedMemorySize,
                            (int)smem_bytes);
  expert_mlp<<<E_ * TILES_, 256, smem_bytes, stream>>>(
      xe, W1, b1, W2, b2, counts, lists, out);
}